// _RPN_71459665871443
// MI455X (gfx1250) — compile-verified
//
#include <hip/hip_runtime.h>

// ---------------------------------------------------------------------------
// Huber box-loss gather + reduction for MI455X (gfx1250, wave32).
//
//   loss = mean over 4*B terms of  (|d| < 3 ? d*d : |d|),
//   d = output[central_pos[b] + 300*b, 4:8] - gt_boxes[b]
//
// Strategy: stage-1 kernel gathers + computes per-image Huber sum, reduces a
// whole wave (32 lanes) in ONE v_wmma_f32_16x16x4_f32 (A = partials striped
// across lanes, B = ones -> D rows are lane-group sums, all in fp32/RNE),
// then LDS-combines 8 waves per block into d_ws. Stage-2 kernel reduces the
// ceil(B/256) block partials the same way and applies the 1/(4B) scale.
// Deterministic tree order -> bit-stable across graph replays.
// ---------------------------------------------------------------------------

#define PROPOSALS_PER_IM 300
#define HUBER_T 3.0f

typedef float v2f __attribute__((ext_vector_type(2)));
typedef float v8f __attribute__((ext_vector_type(8)));

// Sum all 32 lanes' values of a wave using the f32 WMMA unit.
// A (16x4 f32, 2 VGPRs/lane): VGPR0 carries K=0 (lanes 0-15) / K=2 (lanes
// 16-31) = our value; VGPR1 (K=1/K=3) = 0.  B = ones(4x16).
// D[m][n] = v_lane(m) + v_lane(m+16)  (identical for every column n).
// Lane L then sums its 8 D VGPRs: lanes 0-15 hold sum(rows 0-7) = lanes
// {0..7,16..23}; lanes 16-31 hold sum(rows 8-15) = lanes {8..15,24..31}.
// One xor-16 shuffle + add yields the full wave sum in every lane.
__device__ __forceinline__ float wave_sum_wmma(float v) {
    v2f a;
    a[0] = v;
    a[1] = 0.0f;
    v2f bm;
    bm[0] = 1.0f;
    bm[1] = 1.0f;
    v8f c = {};
    // 8 args: (neg_a, A, neg_b, B, c_mod, C, reuse_a, reuse_b)
    v8f d = __builtin_amdgcn_wmma_f32_16x16x4_f32(
        false, a, false, bm, (short)0, c, false, false);
    float s = ((d[0] + d[1]) + (d[2] + d[3])) + ((d[4] + d[5]) + (d[6] + d[7]));
    s += __shfl_xor(s, 16, 32);
    return s;
}

__device__ __forceinline__ float huber4(float4 p, float4 g) {
    float d0 = p.x - g.x, d1 = p.y - g.y, d2 = p.z - g.z, d3 = p.w - g.w;
    float a0 = fabsf(d0), a1 = fabsf(d1), a2 = fabsf(d2), a3 = fabsf(d3);
    float t0 = (a0 < HUBER_T) ? d0 * d0 : a0;
    float t1 = (a1 < HUBER_T) ? d1 * d1 : a1;
    float t2 = (a2 < HUBER_T) ? d2 * d2 : a2;
    float t3 = (a3 < HUBER_T) ? d3 * d3 : a3;
    return (t0 + t1) + (t2 + t3);
}

// Stage 1: one thread per image. 256 threads = 8 waves per block.
__global__ void __launch_bounds__(256)
huber_stage1(const float* __restrict__ out8,  // (B*300, 8)
             const float* __restrict__ gt,    // (B, 4)
             const int* __restrict__ pos,     // (B,)
             float* __restrict__ partials,    // (gridDim.x,)
             int B) {
    __shared__ float lds[8];
    const int i = blockIdx.x * 256 + threadIdx.x;

    float s = 0.0f;
    if (i < B) {
        const long long row = (long long)pos[i] + (long long)PROPOSALS_PER_IM * (long long)i;
        // row*8 + 4 floats -> byte offset row*32 + 16: 16B aligned gather.
        const float4 p = *(const float4*)(out8 + row * 8ll + 4ll);
        const float4 g = *(const float4*)(gt + 4ll * (long long)i);
        s = huber4(p, g);
    }
    // All 256 threads reconverged here: EXEC all-ones for WMMA.
    const float ws = wave_sum_wmma(s);

    const int lane = threadIdx.x & 31;
    const int wave = threadIdx.x >> 5;
    if (lane == 0) lds[wave] = ws;
    __syncthreads();
    if (threadIdx.x == 0) {
        float t = 0.0f;
        #pragma unroll
        for (int w = 0; w < 8; ++w) t += lds[w];
        partials[blockIdx.x] = t;
    }
}

// Stage 2: single 256-thread block folds all block partials and scales.
__global__ void __launch_bounds__(256)
huber_stage2(const float* __restrict__ partials, int n,
             float* __restrict__ out, float inv_count) {
    __shared__ float lds[8];
    float s = 0.0f;
    for (int j = threadIdx.x; j < n; j += 256) s += partials[j];

    const float ws = wave_sum_wmma(s);

    const int lane = threadIdx.x & 31;
    const int wave = threadIdx.x >> 5;
    if (lane == 0) lds[wave] = ws;
    __syncthreads();
    if (threadIdx.x == 0) {
        float t = 0.0f;
        #pragma unroll
        for (int w = 0; w < 8; ++w) t += lds[w];
        out[0] = t * inv_count;
    }
}

extern "C" void kernel_launch(void* const* d_in, const int* in_sizes, int n_in,
                              void* d_out, int out_size, void* d_ws, size_t ws_size,
                              hipStream_t stream) {
    const float* out8 = (const float*)d_in[0];  // (B*300, 8) f32
    const float* gt   = (const float*)d_in[1];  // (B, 4)    f32
    const int*   pos  = (const int*)d_in[2];    // (B,)      int

    const int B = in_sizes[2];
    const int nblocks = (B + 255) / 256;

    float* partials = (float*)d_ws;  // nblocks * 4 bytes of scratch
    float* result   = (float*)d_out;

    huber_stage1<<<nblocks, 256, 0, stream>>>(out8, gt, pos, partials, B);

    const float inv_count = 1.0f / (4.0f * (float)B);
    huber_stage2<<<1, 256, 0, stream>>>(partials, nblocks, result, inv_count);
}